// StackMachineCell_28200755266120
// MI455X (gfx1250) — compile-verified
//
#include <hip/hip_runtime.h>
#include <stdint.h>

#define T_STEPS 512
#define B_SIZE  512
#define H       128
#define NS      64
#define SV      32
#define NMEM    34
#define NBUF    128
#define TCOLS   226   // 34 + 128 + 64 packed logits columns

typedef __attribute__((ext_vector_type(2))) float v2f;
typedef __attribute__((ext_vector_type(8))) float v8f;

// ---------------------------------------------------------------------------
// Precompute: table[v][c] = sum_k embed[v][k] * W[k][c] + b[c]
// One wave per 16x16 tile, V_WMMA_F32_16X16X4_F32, branch-free K-loop
// (clamped load column + select instead of exec-divergent guard).
// ---------------------------------------------------------------------------
__global__ void __launch_bounds__(32)
precompute_tile_kernel(const float* __restrict__ embed,
                       const float* __restrict__ W,
                       const float* __restrict__ bias,
                       float* __restrict__ table,
                       int ncols, int col_off) {
  const int lane  = threadIdx.x & 31;
  const int m0    = blockIdx.x * 16;
  const int n0    = blockIdx.y * 16;
  const int half  = lane >> 4;     // 0: lanes 0-15, 1: lanes 16-31
  const int lid   = lane & 15;
  const int c     = n0 + lid;
  const bool valid = (c < ncols);
  const int cc    = valid ? c : (ncols - 1);   // clamped (always-legal) column

  const float* aptr = embed + (m0 + lid) * H + half * 2;
  const float* bptr = W + (half * 2) * ncols + cc;

  v8f acc = {};
  for (int k = 0; k < H; k += 4) {
    v2f a, b;
    a.x = aptr[0];
    a.y = aptr[1];
    float b0 = bptr[0];
    float b1 = bptr[ncols];
    b.x = valid ? b0 : 0.f;
    b.y = valid ? b1 : 0.f;
    acc = __builtin_amdgcn_wmma_f32_16x16x4_f32(false, a, false, b,
                                                (short)0, acc, false, false);
    aptr += 4;
    bptr += 4 * ncols;
  }
  if (valid) {
    float bv = bias[c];
#pragma unroll
    for (int r = 0; r < 8; ++r) {
      int row = m0 + r + half * 8;
      table[row * TCOLS + col_off + c] = acc[r] + bv;
    }
  }
}

// ---------------------------------------------------------------------------
// CDNA5 async global->LDS DMA (ASYNCcnt-tracked), GVS mode:
//   mem_addr = SADDR(64b sgpr) + VADDR(32b vgpr byte offset), 16B per lane.
// ---------------------------------------------------------------------------
__device__ __forceinline__ void async_ld_b128(unsigned lds_off,
                                              const void* base,
                                              unsigned byte_off) {
  asm volatile("global_load_async_to_lds_b128 %0, %1, %2"
               :
               : "v"(lds_off), "v"(byte_off), "s"(base)
               : "memory");
}

// ---------------------------------------------------------------------------
// Sequential scan: 1 wave32 per batch element, 8 waves/block.
// LDS holds the 96 "one-hot" rows (128..223) of each W (87 KB, CDNA5 WGP LDS).
// Stack (64 ints) lives in registers: lane L owns slots L and L+32.
// ---------------------------------------------------------------------------
__global__ void __launch_bounds__(256)
scan_kernel(const int* __restrict__ x,
            const int* __restrict__ t_act,
            const int* __restrict__ t_s,
            const int* __restrict__ forcing,
            const float* __restrict__ W_mem,
            const float* __restrict__ W_buf,
            const float* __restrict__ W_state,
            const float* __restrict__ table,
            float* __restrict__ lm_out,
            float* __restrict__ lb_out,
            float* __restrict__ ls_out) {
  extern __shared__ float lds[];
  float* Lm = lds;               // [96][34]  rows 128..223 of W_mem
  float* Lb = Lm + 96 * NMEM;    // [96][128]
  float* Ls = Lb + 96 * NBUF;    // [96][64]

  const int tid = threadIdx.x;

  // ---- stage one-hot W rows into LDS via async DMA (b128 per lane) ----
  {
    const unsigned lmo = (unsigned)(uintptr_t)(void*)Lm;
    const unsigned lbo = (unsigned)(uintptr_t)(void*)Lb;
    const unsigned lso = (unsigned)(uintptr_t)(void*)Ls;
    const unsigned t16 = (unsigned)tid * 16u;
    for (unsigned i = t16; i < 96u * NMEM * 4u; i += 256u * 16u)
      async_ld_b128(lmo + i, W_mem, 128u * NMEM * 4u + i);
    for (unsigned i = t16; i < 96u * NBUF * 4u; i += 256u * 16u)
      async_ld_b128(lbo + i, W_buf, 128u * NBUF * 4u + i);
    for (unsigned i = t16; i < 96u * NS * 4u; i += 256u * 16u)
      async_ld_b128(lso + i, W_state, 128u * NS * 4u + i);
    asm volatile("s_wait_asynccnt 0x0" ::: "memory");
  }
  __syncthreads();

  const int lane = tid & 31;
  const int b    = blockIdx.x * 8 + (tid >> 5);

  int ptr = 0, rsym = 0, s = 0;
  int stack_lo = 0, stack_hi = 0;   // slots lane, lane+32

  for (int t = 0; t < T_STEPS; ++t) {
    const int base = t * B_SIZE + b;
    const int xt = x[base];
    const int ta = t_act[base];
    const int ts = t_s[base];
    const int fc = forcing[base];
    const float* trow = table + xt * TCOLS;
    // prefetch next step's table row (x is known ahead; only s,r are recurrent)
    if (t + 1 < T_STEPS) {
      int xn = x[base + B_SIZE];
      __builtin_prefetch(table + xn * TCOLS, 0, 0);
    }
    const int sm = s * NMEM, rm = (64 + rsym) * NMEM;
    const int sb = s * NBUF, rb = (64 + rsym) * NBUF;
    const int ss = s * NS,   rs = (64 + rsym) * NS;

    // ---- mem logits (34) + argmax (first-occurrence tie-break) ----
    float v0 = trow[lane] + Lm[sm + lane] + Lm[rm + lane];
    lm_out[(size_t)base * NMEM + lane] = v0;
    float v1 = -__builtin_inff();
    if (lane < NMEM - 32) {
      int c = lane + 32;
      v1 = trow[c] + Lm[sm + c] + Lm[rm + c];
      lm_out[(size_t)base * NMEM + c] = v1;
    }
    float bestv; int besti;
    if (v1 > v0) { bestv = v1; besti = lane + 32; } else { bestv = v0; besti = lane; }
#pragma unroll
    for (int off = 16; off >= 1; off >>= 1) {
      float ov = __shfl_xor(bestv, off);
      int   oi = __shfl_xor(besti, off);
      if (ov > bestv || (ov == bestv && oi < besti)) { bestv = ov; besti = oi; }
    }
    const int amax_mem = besti;

    // ---- buf logits (128), store only ----
#pragma unroll
    for (int j = 0; j < 4; ++j) {
      int c = lane + 32 * j;
      float v = trow[NMEM + c] + Lb[sb + c] + Lb[rb + c];
      lb_out[(size_t)base * NBUF + c] = v;
    }

    // ---- state logits (64) + argmax ----
    float s0 = trow[NMEM + NBUF + lane] + Ls[ss + lane] + Ls[rs + lane];
    ls_out[(size_t)base * NS + lane] = s0;
    const int c1 = lane + 32;
    float s1 = trow[NMEM + NBUF + c1] + Ls[ss + c1] + Ls[rs + c1];
    ls_out[(size_t)base * NS + c1] = s1;
    float bsv; int bsi;
    if (s1 > s0) { bsv = s1; bsi = c1; } else { bsv = s0; bsi = lane; }
#pragma unroll
    for (int off = 16; off >= 1; off >>= 1) {
      float ov = __shfl_xor(bsv, off);
      int   oi = __shfl_xor(bsi, off);
      if (ov > bsv || (ov == bsv && oi < bsi)) { bsv = ov; bsi = oi; }
    }
    const int amax_state = bsi;

    // ---- recurrence: teacher forcing + register-resident stack ----
    const int action = (fc > 0) ? ta : amax_mem;
    const int s_next = (fc > 0) ? ts : amax_state;
    const int push = (action >= 2) ? 1 : 0;
    const int pop  = (action == 1) ? 1 : 0;
    int np = ptr + push - pop;
    np = np < 0 ? 0 : (np > 63 ? 63 : np);
    int sym = action - 2; if (sym < 0) sym = 0;
    if (push) {
      if (np == lane)      stack_lo = sym;
      if (np == lane + 32) stack_hi = sym;
    }
    const int lo = __shfl(stack_lo, np & 31);
    const int hi = __shfl(stack_hi, np & 31);
    rsym = (np < 32) ? lo : hi;
    ptr = np;
    s = s_next;
  }
}

extern "C" void kernel_launch(void* const* d_in, const int* in_sizes, int n_in,
                              void* d_out, int out_size, void* d_ws, size_t ws_size,
                              hipStream_t stream) {
  const int*   x       = (const int*)d_in[0];
  const int*   ta      = (const int*)d_in[1];
  const int*   ts      = (const int*)d_in[2];
  const int*   fc      = (const int*)d_in[3];
  const float* embed   = (const float*)d_in[4];
  const float* W_mem   = (const float*)d_in[5];
  const float* b_mem   = (const float*)d_in[6];
  const float* W_buf   = (const float*)d_in[7];
  const float* b_buf   = (const float*)d_in[8];
  const float* W_state = (const float*)d_in[9];
  const float* b_state = (const float*)d_in[10];

  float* out   = (float*)d_out;
  float* table = (float*)d_ws;   // 128 * 226 floats = 113 KB scratch

  // table = embed @ W + b  (WMMA f32 16x16x4), packed [mem | buf | state]
  precompute_tile_kernel<<<dim3(8, 3), dim3(32), 0, stream>>>(
      embed, W_mem, b_mem, table, NMEM, 0);
  precompute_tile_kernel<<<dim3(8, 8), dim3(32), 0, stream>>>(
      embed, W_buf, b_buf, table, NBUF, NMEM);
  precompute_tile_kernel<<<dim3(8, 4), dim3(32), 0, stream>>>(
      embed, W_state, b_state, table, NS, NMEM + NBUF);

  float* lm = out;
  float* lb = out + (size_t)T_STEPS * B_SIZE * NMEM;
  float* ls = out + (size_t)T_STEPS * B_SIZE * (NMEM + NBUF);

  const size_t shmem = (size_t)96 * TCOLS * sizeof(float);  // 86,784 B < 320 KB WGP LDS
  scan_kernel<<<dim3(B_SIZE / 8), dim3(256), shmem, stream>>>(
      x, ta, ts, fc, W_mem, W_buf, W_state, table, lm, lb, ls);
}